// StatisticalSMA_2877628089055
// MI455X (gfx1250) — compile-verified
//
#include <hip/hip_runtime.h>
#include <hip/hip_bf16.h>

typedef __attribute__((ext_vector_type(2))) float v2f;
typedef __attribute__((ext_vector_type(8))) float v8f;

#define WINDOW  5
#define OUT_LEN 96
#define SEQ     512
#define CH      64
#define BATCH   2048

// out(96 x 131072) = W(96x5) x C(5x131072), W = linearized SMA recurrence.
// One wave32 per 16x16 output tile, two chained V_WMMA_F32_16X16X4_F32 (K padded 5->8).
__global__ __launch_bounds__(256) void StatisticalSMA_kernel(const float* __restrict__ x,
                                                             float* __restrict__ out) {
    // 96 x 8 coefficient matrix (cols 5..7 zero padding), row-major in LDS.
    __shared__ float As[OUT_LEN * 8];
    const int tid = threadIdx.x;

    // Zero the K-padding columns.
    if (tid < OUT_LEN) {
        As[tid * 8 + 5] = 0.0f;
        As[tid * 8 + 6] = 0.0f;
        As[tid * 8 + 7] = 0.0f;
    }
    // Threads 0..4: simulate recurrence on basis vector e_j to get coefficient column j.
    if (tid < WINDOW) {
        float c0 = (tid == 0) ? 1.0f : 0.0f;
        float c1 = (tid == 1) ? 1.0f : 0.0f;
        float c2 = (tid == 2) ? 1.0f : 0.0f;
        float c3 = (tid == 3) ? 1.0f : 0.0f;
        float c4 = (tid == 4) ? 1.0f : 0.0f;
        for (int t = 0; t < OUT_LEN; ++t) {
            float f = (c0 + c1 + c2 + c3 + c4) * 0.2f;
            As[t * 8 + tid] = f;
            c0 = c1; c1 = c2; c2 = c3; c3 = c4; c4 = f;
        }
    }
    __syncthreads();

    const int wave    = tid >> 5;
    const int lane    = tid & 31;
    const int lane_lo = lane & 15;
    const int khalf   = lane >> 4;          // 0: lanes 0-15, 1: lanes 16-31

    // Global wave id -> (m_tile in [0,6), n_tile in [0,8192))
    const int g      = blockIdx.x * 8 + wave;
    const int m_tile = g % 6;
    const int n_tile = g / 6;
    const int M0     = m_tile * 16;
    const int n      = n_tile * 16 + lane_lo;   // flattened (b,ch) column
    const int b      = n >> 6;
    const int ch     = n & 63;

    // A fragments (16x4 f32 layout: VGPR0 -> K = 2*khalf, VGPR1 -> K = 2*khalf+1)
    const float* Arow = &As[(M0 + lane_lo) * 8];
    v2f a0, a1;
    a0.x = Arow[2 * khalf + 0];
    a0.y = Arow[2 * khalf + 1];
    a1.x = Arow[4 + 2 * khalf + 0];
    a1.y = Arow[4 + 2 * khalf + 1];

    // B fragments: B[k][n] = x[b, 507+k, ch]; rows 5..7 are zero.
    const float* xb = x + (size_t)b * SEQ * CH + ch;
    const int r0 = (SEQ - WINDOW) + 2 * khalf;          // 507 or 509 (in bounds)
    const float bv0 = xb[(size_t)r0 * CH];              // k = 2*khalf
    const float bv1 = xb[(size_t)(r0 + 1) * CH];        // k = 2*khalf+1
    const float bv4 = xb[(size_t)(SEQ - 1) * CH];       // row 511 = k=4 (always in bounds)
    v2f b0, b1;
    b0.x = bv0;
    b0.y = bv1;
    b1.x = (khalf == 0) ? bv4 : 0.0f;                   // k=4 real, k=6 padded zero
    b1.y = 0.0f;                                        // k=5 / k=7 padded zero

    v8f acc = {};
    acc = __builtin_amdgcn_wmma_f32_16x16x4_f32(false, a0, false, b0, (short)0, acc, false, false);
    acc = __builtin_amdgcn_wmma_f32_16x16x4_f32(false, a1, false, b1, (short)0, acc, false, false);

    // D layout: VGPR v -> t = M0 + v + 8*khalf, column n (16 consecutive ch per half-wave).
    float* op = out + (size_t)b * OUT_LEN * CH + ch;
#pragma unroll
    for (int v = 0; v < 8; ++v) {
        const int t = M0 + v + 8 * khalf;
        op[(size_t)t * CH] = acc[v];
    }
}

extern "C" void kernel_launch(void* const* d_in, const int* in_sizes, int n_in,
                              void* d_out, int out_size, void* d_ws, size_t ws_size,
                              hipStream_t stream) {
    (void)in_sizes; (void)n_in; (void)d_ws; (void)ws_size; (void)out_size;
    const float* x = (const float*)d_in[0];
    float* out = (float*)d_out;
    // 6 m-tiles * 8192 n-tiles = 49152 waves / 8 waves per 256-thread block = 6144 blocks
    StatisticalSMA_kernel<<<6144, 256, 0, stream>>>(x, out);
}